// RADecay_31361851195436
// MI455X (gfx1250) — compile-verified
//
#include <hip/hip_runtime.h>
#include <hip/hip_bf16.h>
#include <math.h>

typedef __attribute__((ext_vector_type(2))) float v2f;
typedef __attribute__((ext_vector_type(4))) float v4f;
typedef __attribute__((ext_vector_type(8))) float v8f;

#define ABLK 256            // threads per block in fused kernel (8 waves)
#define RPB  1024           // rows per block (fused kernel)
#define KTOP 64
#define MERGE_T 1024
#define MAXC 32768          // max candidate count supported by merge bitmask
#define LN_DECAY -0.10536051565782628f   // ln(0.9)

__device__ __forceinline__ float wave_reduce_add(float v) {
    v += __shfl_xor(v, 16, 32);
    v += __shfl_xor(v, 8, 32);
    v += __shfl_xor(v, 4, 32);
    v += __shfl_xor(v, 2, 32);
    v += __shfl_xor(v, 1, 32);
    return v;
}

// ---------------------------------------------------------------------------
// Kernel A: fused  (read fs once) -> copy fs to new_fs (NT), alpha dot,
//           per-block local top-64 extraction from LDS.
// One wave per row: 32 lanes x float4 == 128 floats (perfectly coalesced).
// ---------------------------------------------------------------------------
__global__ __launch_bounds__(ABLK) void fused_alpha_copy_topk(
    const float* __restrict__ fs, const float* __restrict__ feature,
    float* __restrict__ out_fs, float* __restrict__ cand_val,
    int* __restrict__ cand_idx, int N)
{
    __shared__ float s_alpha[RPB];
    __shared__ float s_redv[ABLK];
    __shared__ int   s_redi[ABLK];

    const int tid  = threadIdx.x;
    const int lane = tid & 31;
    const int wave = tid >> 5;
    const int rowBase = blockIdx.x * RPB;

    const v4f f4 = ((const v4f*)feature)[lane];   // lane-private slice, loaded once

    for (int r = wave; r < RPB; r += (ABLK / 32)) {
        const int row = rowBase + r;
        if (row < N) {
            const v4f* src = (const v4f*)(fs + (size_t)row * 128) + lane;
            v4f v = __builtin_nontemporal_load(src);
            v4f* dst = (v4f*)(out_fs + (size_t)row * 128) + lane;
            __builtin_nontemporal_store(v, dst);
            float dot = v.x * f4.x + v.y * f4.y + v.z * f4.z + v.w * f4.w;
            dot = wave_reduce_add(dot);
            if (row + 64 < N)
                __builtin_prefetch(fs + (size_t)(row + 64) * 128 + lane * 4, 0, 0);
            if (lane == 0) s_alpha[r] = dot;
        } else {
            if (lane == 0) s_alpha[r] = -__builtin_inff();
        }
    }
    __syncthreads();

    // extract block-local top-64 (argmax 64 times over LDS)
    for (int it = 0; it < KTOP; ++it) {
        float bv = -__builtin_inff(); int bi = 0;
#pragma unroll
        for (int j = 0; j < RPB / ABLK; ++j) {
            const int i = tid + j * ABLK;
            const float v = s_alpha[i];
            if (v > bv) { bv = v; bi = i; }
        }
        s_redv[tid] = bv; s_redi[tid] = bi;
        __syncthreads();
        for (int s = ABLK / 2; s > 0; s >>= 1) {
            if (tid < s) {
                if (s_redv[tid + s] > s_redv[tid]) {
                    s_redv[tid] = s_redv[tid + s];
                    s_redi[tid] = s_redi[tid + s];
                }
            }
            __syncthreads();
        }
        if (tid == 0) {
            cand_val[blockIdx.x * KTOP + it] = s_redv[0];
            cand_idx[blockIdx.x * KTOP + it] = rowBase + s_redi[0];
            s_alpha[s_redi[0]] = -__builtin_inff();
        }
        __syncthreads();
    }
}

// ---------------------------------------------------------------------------
// Kernel A2: streaming NT copy of hs -> new_hs (float4) and ts -> new_ts.
// ---------------------------------------------------------------------------
__global__ __launch_bounds__(256) void copy_hs_ts(
    const float* __restrict__ hs, const float* __restrict__ ts,
    float* __restrict__ out_hs, float* __restrict__ out_ts, int N)
{
    const size_t n4 = (size_t)N * 32;          // float4 count of hs
    const size_t total = n4 + (size_t)N;
    for (size_t i = (size_t)blockIdx.x * blockDim.x + threadIdx.x; i < total;
         i += (size_t)gridDim.x * blockDim.x) {
        if (i < n4) {
            v4f v = __builtin_nontemporal_load(((const v4f*)hs) + i);
            __builtin_nontemporal_store(v, ((v4f*)out_hs) + i);
        } else {
            const size_t j = i - n4;
            out_ts[j] = ts[j];
        }
    }
}

// ---------------------------------------------------------------------------
// Kernel B: merge block-local candidates (C = nb*64) into global top-64.
// Used-mask kept in LDS so __syncthreads fully orders it.
// ---------------------------------------------------------------------------
__global__ __launch_bounds__(MERGE_T) void merge_topk(
    const float* __restrict__ cand_val, const int* __restrict__ cand_idx,
    float* __restrict__ topv, int* __restrict__ topi, int C)
{
    __shared__ float    s_redv[MERGE_T];
    __shared__ int      s_redi[MERGE_T];
    __shared__ unsigned s_used[MAXC / 32];

    const int tid = threadIdx.x;
    for (int i = tid; i < MAXC / 32; i += MERGE_T) s_used[i] = 0u;
    __syncthreads();

    for (int it = 0; it < KTOP; ++it) {
        float bv = -__builtin_inff(); int bi = -1;
        for (int i = tid; i < C; i += MERGE_T) {
            if ((s_used[i >> 5] >> (i & 31)) & 1u) continue;
            const float v = cand_val[i];
            if (v > bv) { bv = v; bi = i; }
        }
        s_redv[tid] = bv; s_redi[tid] = bi;
        __syncthreads();
        for (int s = MERGE_T / 2; s > 0; s >>= 1) {
            if (tid < s) {
                if (s_redv[tid + s] > s_redv[tid]) {
                    s_redv[tid] = s_redv[tid + s];
                    s_redi[tid] = s_redi[tid + s];
                }
            }
            __syncthreads();
        }
        if (tid == 0) {
            const int b = s_redi[0];
            topv[it] = s_redv[0];
            topi[it] = (b >= 0) ? cand_idx[b] : 0;
            if (b >= 0) s_used[b >> 5] |= (1u << (b & 31));
        }
        __syncthreads();
    }
}

// ---------------------------------------------------------------------------
// Kernel C: decay + softmax + attn gather + GRU (WMMA f32 16x16x4) +
//           output projection + log_softmax + append rows. One block, 8 waves.
// ---------------------------------------------------------------------------
__global__ __launch_bounds__(256) void finalize_kernel(
    const float* __restrict__ feature, const float* __restrict__ time_p,
    const float* __restrict__ hs, const float* __restrict__ ts,
    const float* __restrict__ w_ih, const float* __restrict__ w_hh,
    const float* __restrict__ b_ih, const float* __restrict__ b_hh,
    const float* __restrict__ w_out, const float* __restrict__ b_out,
    const float* __restrict__ topv, const int* __restrict__ topi,
    float* __restrict__ out_pred, float* __restrict__ out_fs,
    float* __restrict__ out_hs, float* __restrict__ out_ts, int N)
{
    __shared__ float s_feat[128], s_h[128], s_attn[128];
    __shared__ float s_a[64];
    __shared__ int   s_i[64];
    __shared__ float s_gi[384], s_gh[384];
    __shared__ float s_pred[64];
    __shared__ float s_scalar[2];

    const int tid = threadIdx.x;
    const float tnow = time_p[0];

    if (tid < 128) {
        const float fv = feature[tid];
        s_feat[tid] = fv;
        out_fs[(size_t)N * 128 + tid] = fv;                 // append feature row
        s_h[tid] = hs[(size_t)(N - 1) * 128 + tid];         // last hidden
    }
    if (tid == 0) out_ts[N] = tnow;                         // append timestamp
    if (tid < 64) {
        const int idx = topi[tid];
        s_i[tid] = idx;
        const float dt = tnow - ts[idx];
        s_a[tid] = topv[tid] * __expf(dt * LN_DECAY);       // 0.9^dt decay
    }
    __syncthreads();

    // softmax over the 64 decayed scores
    if (tid == 0) {
        float m = -__builtin_inff();
        for (int k = 0; k < 64; ++k) m = fmaxf(m, s_a[k]);
        s_scalar[0] = m;
    }
    __syncthreads();
    if (tid < 64) s_a[tid] = __expf(s_a[tid] - s_scalar[0]);
    __syncthreads();
    if (tid == 0) {
        float s = 0.f;
        for (int k = 0; k < 64; ++k) s += s_a[k];
        s_scalar[1] = 1.0f / s;
    }
    __syncthreads();
    if (tid < 64) s_a[tid] *= s_scalar[1];
    __syncthreads();

    // attn_h = a @ hs[idx]  (coalesced gather across 128 lanes)
    if (tid < 128) {
        float acc = 0.f;
        for (int k = 0; k < 64; ++k)
            acc += s_a[k] * hs[(size_t)s_i[k] * 128 + tid];
        s_attn[tid] = acc;
    }
    __syncthreads();

    // --- GRU gate matvecs via V_WMMA_F32_16X16X4_F32 -----------------------
    // gi = w_ih@feature, gh = w_hh@h.  48 row-tiles of 16, 6 per wave.
    // A layout (16x4 f32): lane m=lane&15 holds row, vgpr pair = K {2*half, 2*half+1}.
    // B column 0 carries the vector (lanes with N==0), other columns zero.
    {
        const int lane = tid & 31;
        const int wave = tid >> 5;
        const int m    = lane & 15;
        const int half = lane >> 4;
        for (int j = 0; j < 6; ++j) {
            const int t = wave * 6 + j;
            const bool isIH = (t < 24);
            const float* W    = isIH ? w_ih : w_hh;
            const float* vec  = isIH ? s_feat : s_h;
            const float* bias = isIH ? b_ih : b_hh;
            float* dst        = isIH ? s_gi : s_gh;
            const int rowBase = (isIH ? t : t - 24) * 16;

            v8f acc = {};
            for (int k = 0; k < 128; k += 4) {
                const v2f a =
                    *(const v2f*)(W + (size_t)(rowBase + m) * 128 + k + 2 * half);
                v2f b;
                b.x = (m == 0) ? vec[k + 2 * half]     : 0.0f;
                b.y = (m == 0) ? vec[k + 2 * half + 1] : 0.0f;
                acc = __builtin_amdgcn_wmma_f32_16x16x4_f32(
                    false, a, false, b, (short)0, acc, false, false);
            }
            if (m == 0) {   // D column 0 lives in lanes 0 (rows 0-7) / 16 (rows 8-15)
#pragma unroll
                for (int r = 0; r < 8; ++r) {
                    const int row = rowBase + 8 * half + r;
                    dst[row] = acc[r] + bias[row];
                }
            }
        }
    }
    __syncthreads();

    // GRU combine -> new_h (appended row of new_hs)
    if (tid < 128) {
        const float r = 1.f / (1.f + __expf(-(s_gi[tid] + s_gh[tid])));
        const float z = 1.f / (1.f + __expf(-(s_gi[128 + tid] + s_gh[128 + tid])));
        const float n = tanhf(s_gi[256 + tid] + r * s_gh[256 + tid]);
        out_hs[(size_t)N * 128 + tid] = (1.f - z) * n + z * s_h[tid];
    }

    // pred = w_out @ [feature, attn_h, h, K] + b_out  (64 x 385 matvec)
    if (tid < 64) {
        const float* wrow = w_out + (size_t)tid * 385;
        float acc = b_out[tid];
        for (int j = 0; j < 128; ++j) acc += wrow[j]       * s_feat[j];
        for (int j = 0; j < 128; ++j) acc += wrow[128 + j] * s_attn[j];
        for (int j = 0; j < 128; ++j) acc += wrow[256 + j] * s_h[j];
        acc += wrow[384] * 64.0f;                           // length = K
        s_pred[tid] = acc;
    }
    __syncthreads();

    // log_softmax
    if (tid == 0) {
        float m = -__builtin_inff();
        for (int k = 0; k < 64; ++k) m = fmaxf(m, s_pred[k]);
        float s = 0.f;
        for (int k = 0; k < 64; ++k) s += __expf(s_pred[k] - m);
        s_scalar[0] = m; s_scalar[1] = logf(s);
    }
    __syncthreads();
    if (tid < 64) out_pred[tid] = s_pred[tid] - s_scalar[0] - s_scalar[1];
}

// ---------------------------------------------------------------------------
extern "C" void kernel_launch(void* const* d_in, const int* in_sizes, int n_in,
                              void* d_out, int out_size, void* d_ws, size_t ws_size,
                              hipStream_t stream)
{
    (void)n_in; (void)out_size; (void)ws_size;
    const float* feature = (const float*)d_in[0];
    const float* time_p  = (const float*)d_in[1];
    const float* fs      = (const float*)d_in[2];
    const float* hs      = (const float*)d_in[3];
    const float* ts      = (const float*)d_in[4];
    const float* w_ih    = (const float*)d_in[5];
    const float* w_hh    = (const float*)d_in[6];
    const float* b_ih    = (const float*)d_in[7];
    const float* b_hh    = (const float*)d_in[8];
    const float* w_out   = (const float*)d_in[9];
    const float* b_out   = (const float*)d_in[10];

    const int N  = in_sizes[2] / 128;
    const int nb = (N + RPB - 1) / RPB;
    const int C  = nb * KTOP;

    float* out_pred = (float*)d_out;
    float* out_fs   = out_pred + 64;
    float* out_hs   = out_fs + (size_t)(N + 1) * 128;
    float* out_ts   = out_hs + (size_t)(N + 1) * 128;

    float* cand_val = (float*)d_ws;
    int*   cand_idx = (int*)(cand_val + C);
    float* topv     = (float*)(cand_idx + C);
    int*   topi     = (int*)(topv + KTOP);

    fused_alpha_copy_topk<<<nb, ABLK, 0, stream>>>(fs, feature, out_fs,
                                                   cand_val, cand_idx, N);

    const size_t total = (size_t)N * 32 + (size_t)N;
    int cblocks = (int)((total + 255) / 256);
    if (cblocks > 16384) cblocks = 16384;
    copy_hs_ts<<<cblocks, 256, 0, stream>>>(hs, ts, out_hs, out_ts, N);

    merge_topk<<<1, MERGE_T, 0, stream>>>(cand_val, cand_idx, topv, topi, C);

    finalize_kernel<<<1, 256, 0, stream>>>(feature, time_p, hs, ts,
                                           w_ih, w_hh, b_ih, b_hh, w_out, b_out,
                                           topv, topi, out_pred, out_fs, out_hs,
                                           out_ts, N);
}